// DMTetGeometry_23648089932440
// MI455X (gfx1250) — compile-verified
//
#include <hip/hip_runtime.h>
#include <stdint.h>

// ---------------------------------------------------------------------------
// DMTet (marching tetrahedra) for MI455X / gfx1250.  Data-movement bound, no
// matmul.  CDNA5-specific paths:
//   * wave32 ballot -> 1-bit occupancy mask (50 KB, WGP$-resident gathers)
//   * GLOBAL_LOAD_ASYNC_TO_LDS_B128 double-buffered (ASYNCcnt) in classify:
//     issue tile i+1, s_wait_asynccnt 1, consume tile i -- barrier-free since
//     each lane reads back only its own staged row.
//   * global_prefetch_b8 via __builtin_prefetch, TH_STORE_NT on write-once
//     output sections (everything else is L2-resident: total footprint ~40MB
//     vs 192MB L2, so passes are L2-bandwidth bound, not HBM).
// Surviving tet rows are compacted once (validRow/innerRow) so all later
// passes are contiguous b128 streams instead of random gathers.
// ---------------------------------------------------------------------------

struct Counters {
  int numValid;   // surface tets (0 < occ_sum < 4)
  int numInner;   // occ_sum == 4 tets
  int numCross;   // E: unique crossing edges (interp verts)
  int numFaces;   // emitted triangles
  int numUniq;    // U: unique vertex ids in output tet mesh
  int tetWrite;   // cursor for emitted tets
  int pad0, pad1;
};

__device__ __constant__ signed char c_TRI[16][6] = {
  {-1,-1,-1,-1,-1,-1},{1,0,2,-1,-1,-1},{4,0,3,-1,-1,-1},{1,4,2,1,3,4},
  {3,1,5,-1,-1,-1},{2,3,0,2,5,3},{1,4,0,1,5,4},{4,2,5,-1,-1,-1},
  {4,5,2,-1,-1,-1},{4,1,0,4,5,1},{3,2,0,3,5,2},{1,3,5,-1,-1,-1},
  {4,1,2,4,3,1},{3,0,4,-1,-1,-1},{2,0,1,-1,-1,-1},{-1,-1,-1,-1,-1,-1}};
__device__ __constant__ signed char c_NTRI[16]  = {0,1,1,2,1,2,2,1,1,2,2,1,2,1,1,0};
__device__ __constant__ signed char c_NTET[16]  = {0,1,1,3,1,3,3,3,1,3,3,3,3,3,3,1};
__device__ __constant__ signed char c_EA[6] = {0,0,0,1,1,2};
__device__ __constant__ signed char c_EB[6] = {1,2,3,2,3,3};
__device__ __constant__ signed char c_TET[16][12] = {
  {-1,-1,-1,-1,-1,-1,-1,-1,-1,-1,-1,-1},{0,4,5,6,-1,-1,-1,-1,-1,-1,-1,-1},
  {1,4,8,7,-1,-1,-1,-1,-1,-1,-1,-1},{7,1,8,6,5,1,7,6,5,0,1,6},
  {2,5,7,9,-1,-1,-1,-1,-1,-1,-1,-1},{4,0,6,7,9,0,7,6,7,0,9,2},
  {4,1,9,8,5,1,9,4,5,1,2,9},{6,0,1,2,8,6,1,2,9,6,8,2},
  {3,6,9,8,-1,-1,-1,-1,-1,-1,-1,-1},{5,0,4,8,5,0,8,3,5,8,9,3},
  {1,4,7,3,4,7,6,3,9,6,7,3},{0,1,5,3,5,1,9,3,5,1,7,9},
  {5,2,3,7,3,6,5,8,3,5,7,8},{0,4,7,8,0,3,8,7,0,3,7,2},
  {4,1,2,3,4,3,2,5,4,3,5,6},{0,1,2,3,-1,-1,-1,-1,-1,-1,-1,-1}};

static constexpr unsigned long long EMPTY_KEY = ~0ull;

static __device__ __forceinline__ int occAt(const uint32_t* __restrict__ ob, int v) {
  return (int)((ob[v >> 5] >> (v & 31)) & 1u);
}
static __device__ __forceinline__ int tcomp(int4 t, int c) {
  return c == 0 ? t.x : (c == 1 ? t.y : (c == 2 ? t.z : t.w));
}
static __device__ __forceinline__ uint64_t mix64(uint64_t x) {
  x ^= x >> 33; x *= 0xff51afd7ed558ccdULL;
  x ^= x >> 33; x *= 0xc4ceb9fe1a85ec53ULL;
  x ^= x >> 33; return x;
}
static __device__ __forceinline__ void ntStore(float* p, float v) {
  __builtin_nontemporal_store(v, p);     // th:TH_STORE_NT -- write-once output
}

// --- K1: occupancy bitmask via wave32 ballot --------------------------------
__global__ void k_occ(const float* __restrict__ sdf, int n,
                      const float* __restrict__ thick, uint32_t* __restrict__ occBits) {
  float th = thick[0];
  int i = blockIdx.x * blockDim.x + threadIdx.x;
  bool pred = false;
  if (i < n) { float s = sdf[i]; pred = (s > 0.f) && (s <= th); }
#if defined(__gfx1250__) && __has_builtin(__builtin_amdgcn_ballot_w32)
  uint32_t m = __builtin_amdgcn_ballot_w32(pred);
  if ((threadIdx.x & 31) == 0) {
    int w = i >> 5;
    if ((w << 5) < n) occBits[w] = m;
  }
#else
  if (i < n) {
    if ((i & 31) == 0) occBits[i >> 5] = 0u;
    __syncthreads();
    if (pred) atomicOr(&occBits[i >> 5], 1u << (i & 31));
  }
#endif
}

// --- K2: classify tets.  Double-buffered ASYNCcnt DMA staging: each lane
//     stages its own 16B tet row to LDS; per-wave in-order completion means
//     s_wait_asynccnt 1 retires the older tile -- no barriers needed. -------
__global__ void k_classify(const int* __restrict__ tet, int F,
                           const uint32_t* __restrict__ occBits,
                           Counters* cnt,
                           int4* __restrict__ validRow, int* __restrict__ validTi,
                           int4* __restrict__ innerRow) {
  const int tid = threadIdx.x;
  const long long stride = (long long)gridDim.x * 256;
  long long base = (long long)blockIdx.x * 256;
  if (base >= F) return;
#if defined(__gfx1250__)
  __shared__ int4 tile[2][256];
  auto issue = [&](long long b, int buf) {
    long long f = b + tid; if (f >= F) f = F - 1;
    uint32_t lds = (uint32_t)(uintptr_t)(&tile[buf][tid]);       // LDS byte addr
    uint64_t ga  = (uint64_t)(uintptr_t)(tet + f * 4);
    asm volatile("global_load_async_to_lds_b128 %0, %1, off"
                 :: "v"(lds), "v"(ga) : "memory");
  };
  int buf = 0;
  issue(base, buf);
  for (; base < F; base += stride) {
    long long nxt = base + stride;
    if (nxt < F) {
      issue(nxt, buf ^ 1);
      asm volatile("s_wait_asynccnt 1" ::: "memory");   // older tile landed
    } else {
      asm volatile("s_wait_asynccnt 0" ::: "memory");
    }
    int4 t = tile[buf][tid];                            // self-staged row only
    buf ^= 1;
#else
  for (; base < F; base += stride) {
    long long fc = base + tid < F ? base + tid : F - 1;
    int4 t = ((const int4*)tet)[fc];
#endif
    if (base + tid < F) {
      int o0 = occAt(occBits, t.x), o1 = occAt(occBits, t.y);
      int o2 = occAt(occBits, t.z), o3 = occAt(occBits, t.w);
      int ti = o0 | (o1 << 1) | (o2 << 2) | (o3 << 3);
      int s  = o0 + o1 + o2 + o3;
      if (s == 4) {
        int k = atomicAdd(&cnt->numInner, 1);
        innerRow[k] = t;                                // compact once,
      } else if (s > 0) {                               // stream later
        int k = atomicAdd(&cnt->numValid, 1);
        validRow[k] = t;
        validTi[k]  = ti;
      }
    }
  }
}

// --- K3a: insert sorted edges of surface tets into u64 hash (dedupe) --------
__global__ void k_edge_insert(const int4* __restrict__ validRow, const Counters* cnt,
                              unsigned long long* __restrict__ hkey, unsigned hmask) {
  int V = cnt->numValid;
  int stride = gridDim.x * blockDim.x;
  for (int v = blockIdx.x * blockDim.x + threadIdx.x; v < V; v += stride) {
    __builtin_prefetch(&validRow[v + 4 * stride], 0, 0);   // global_prefetch_b8
    int4 t = validRow[v];
#pragma unroll
    for (int e = 0; e < 6; ++e) {
      int a = tcomp(t, (int)c_EA[e]);
      int b = tcomp(t, (int)c_EB[e]);
      if (a > b) { int tmp = a; a = b; b = tmp; }
      unsigned long long key = ((unsigned long long)(uint32_t)a << 32) | (uint32_t)b;
      unsigned slot = (unsigned)mix64(key) & hmask;
      for (unsigned p = 0; p <= hmask; ++p) {
        unsigned long long old = atomicCAS(&hkey[slot], EMPTY_KEY, key);
        if (old == EMPTY_KEY || old == key) break;
        slot = (slot + 1) & hmask;
      }
    }
  }
}

// --- K3b: scan hash slots, assign compact ids to crossing edges -------------
__global__ void k_edge_assign(const unsigned long long* __restrict__ hkey,
                              int* __restrict__ hval, unsigned hsize,
                              const uint32_t* __restrict__ occBits, Counters* cnt,
                              int* __restrict__ interpA, int* __restrict__ interpB) {
  unsigned stride = gridDim.x * blockDim.x;
  for (unsigned s = blockIdx.x * blockDim.x + threadIdx.x; s < hsize; s += stride) {
    unsigned long long k = hkey[s];
    if (k == EMPTY_KEY) { hval[s] = -1; continue; }
    int a = (int)(k >> 32), b = (int)(uint32_t)k;
    if (occAt(occBits, a) ^ occAt(occBits, b)) {        // exactly one inside band
      int e = atomicAdd(&cnt->numCross, 1);
      hval[s] = e; interpA[e] = a; interpB[e] = b;
    } else {
      hval[s] = -1;
    }
  }
}

// --- K3c: per-tet hash lookups -> idx_map -----------------------------------
__global__ void k_idxmap(const int4* __restrict__ validRow, const Counters* cnt,
                         const unsigned long long* __restrict__ hkey,
                         const int* __restrict__ hval, unsigned hmask,
                         int* __restrict__ idxMap) {
  int V = cnt->numValid;
  int stride = gridDim.x * blockDim.x;
  for (int v = blockIdx.x * blockDim.x + threadIdx.x; v < V; v += stride) {
    __builtin_prefetch(&validRow[v + 4 * stride], 0, 0);
    int4 t = validRow[v];
#pragma unroll
    for (int e = 0; e < 6; ++e) {
      int a = tcomp(t, (int)c_EA[e]);
      int b = tcomp(t, (int)c_EB[e]);
      if (a > b) { int tmp = a; a = b; b = tmp; }
      unsigned long long key = ((unsigned long long)(uint32_t)a << 32) | (uint32_t)b;
      unsigned slot = (unsigned)mix64(key) & hmask;
      int val = -1;
      for (unsigned p = 0; p <= hmask; ++p) {
        unsigned long long kk = hkey[slot];
        if (kk == key) { val = hval[slot]; break; }
        if (kk == EMPTY_KEY) break;
        slot = (slot + 1) & hmask;
      }
      idxMap[(size_t)v * 6 + e] = val;
    }
  }
}

// --- K4: edge interpolation -> verts (d_out section 0, re-read by k_vtm) ----
__global__ void k_interp(const float* __restrict__ pos, const float* __restrict__ sdf,
                         const float* __restrict__ thick, const Counters* cnt,
                         const int* __restrict__ interpA, const int* __restrict__ interpB,
                         float* __restrict__ out, long long outCap) {
  float th = thick[0];
  int E = cnt->numCross;
  int stride = gridDim.x * blockDim.x;
  for (int e = blockIdx.x * blockDim.x + threadIdx.x; e < E; e += stride) {
    int a = interpA[e], b = interpB[e];
    float s0 = sdf[a], s1 = sdf[b];
    if (s0 > 0.f && s1 > 0.f) { s0 -= th; s1 -= th; }   // both inside +band shift
    float denom = s0 - s1;                              // s0 + (-s1)
    float w0 = (-s1) / denom, w1 = s0 / denom;          // flip/denom
    long long o = (long long)e * 3;
    if (o + 2 < outCap) {
      out[o + 0] = pos[(size_t)a * 3 + 0] * w0 + pos[(size_t)b * 3 + 0] * w1;
      out[o + 1] = pos[(size_t)a * 3 + 1] * w0 + pos[(size_t)b * 3 + 1] * w1;
      out[o + 2] = pos[(size_t)a * 3 + 2] * w0 + pos[(size_t)b * 3 + 2] * w1;
    }
  }
}

// --- K5: emit triangles (d_out section 1), NT stores ------------------------
__global__ void k_faces(Counters* cnt, const int* __restrict__ validTi,
                        const int* __restrict__ idxMap,
                        float* __restrict__ out, long long outCap) {
  int V = cnt->numValid;
  long long base = (long long)cnt->numCross * 3;
  int stride = gridDim.x * blockDim.x;
  for (int v = blockIdx.x * blockDim.x + threadIdx.x; v < V; v += stride) {
    int ti = validTi[v];
    int nt = (int)c_NTRI[ti];
    for (int k = 0; k < nt; ++k) {
      int slot = atomicAdd(&cnt->numFaces, 1);
      long long o = base + (long long)slot * 3;
      if (o + 2 < outCap) {
        ntStore(&out[o + 0], (float)idxMap[(size_t)v * 6 + (int)c_TRI[ti][k * 3 + 0]]);
        ntStore(&out[o + 1], (float)idxMap[(size_t)v * 6 + (int)c_TRI[ti][k * 3 + 1]]);
        ntStore(&out[o + 2], (float)idxMap[(size_t)v * 6 + (int)c_TRI[ti][k * 3 + 2]]);
      }
    }
  }
}

// --- K6: mark all vertex ids referenced by the output tet mesh --------------
__global__ void k_mark(const Counters* cnt,
                       const int4* __restrict__ validRow, const int* __restrict__ validTi,
                       const int4* __restrict__ innerRow, const int* __restrict__ idxMap,
                       int N, unsigned char* __restrict__ flags) {
  int V = cnt->numValid, I = cnt->numInner;
  int stride = gridDim.x * blockDim.x;
  for (int idx = blockIdx.x * blockDim.x + threadIdx.x; idx < V + I; idx += stride) {
    if (idx < V) {
      int4 t = validRow[idx];
      int ti = validTi[idx];
      int ntt = (int)c_NTET[ti];
      for (int k = 0; k < ntt; ++k)
        for (int c = 0; c < 4; ++c) {
          int e = (int)c_TET[ti][k * 4 + c];
          int id = (e < 4) ? tcomp(t, e) : (N + idxMap[(size_t)idx * 6 + (e - 4)]);
          flags[id] = 1;
        }
    } else {
      int4 t = innerRow[idx - V];
      flags[t.x] = 1; flags[t.y] = 1; flags[t.z] = 1; flags[t.w] = 1;
    }
  }
}

// --- K7: compact flagged ids -> unique list + remap table -------------------
__global__ void k_compact(Counters* cnt, const unsigned char* __restrict__ flags,
                          int* __restrict__ remap, int* __restrict__ uniqL, int N) {
  long long bound = (long long)N + cnt->numCross;
  long long stride = (long long)gridDim.x * blockDim.x;
  for (long long id = blockIdx.x * (long long)blockDim.x + threadIdx.x;
       id < bound; id += stride) {
    if (flags[id]) {
      int u = atomicAdd(&cnt->numUniq, 1);
      remap[id] = u;
      uniqL[u] = (int)id;
    }
  }
}

// --- K8: gather verts_tetmesh (d_out section 2), NT stores ------------------
__global__ void k_vtm(const float* __restrict__ pos, const Counters* cnt,
                      const int* __restrict__ uniqL, int N,
                      float* __restrict__ out, long long outCap) {
  int U = cnt->numUniq;
  long long base = (long long)(cnt->numCross + cnt->numFaces) * 3;
  int stride = gridDim.x * blockDim.x;
  for (int u = blockIdx.x * blockDim.x + threadIdx.x; u < U; u += stride) {
    int id = uniqL[u];
    long long o = base + (long long)u * 3;
    if (o + 2 < outCap) {
      if (id < N) {
        ntStore(&out[o + 0], pos[(size_t)id * 3 + 0]);
        ntStore(&out[o + 1], pos[(size_t)id * 3 + 1]);
        ntStore(&out[o + 2], pos[(size_t)id * 3 + 2]);
      } else {
        long long e = (long long)(id - N) * 3;   // interp verts live at d_out[0..3E)
        ntStore(&out[o + 0], out[e + 0]);
        ntStore(&out[o + 1], out[e + 1]);
        ntStore(&out[o + 2], out[e + 2]);
      }
    }
  }
}

// --- K9: emit remapped tets (d_out section 3), NT stores --------------------
__global__ void k_tets(Counters* cnt,
                       const int4* __restrict__ validRow, const int* __restrict__ validTi,
                       const int4* __restrict__ innerRow, const int* __restrict__ idxMap,
                       const int* __restrict__ remap, int N,
                       float* __restrict__ out, long long outCap) {
  int V = cnt->numValid, I = cnt->numInner;
  long long base = (long long)(cnt->numCross + cnt->numFaces + cnt->numUniq) * 3;
  int stride = gridDim.x * blockDim.x;
  for (int idx = blockIdx.x * blockDim.x + threadIdx.x; idx < V + I; idx += stride) {
    if (idx < V) {
      int4 t = validRow[idx];
      int ti = validTi[idx];
      int ntt = (int)c_NTET[ti];
      for (int k = 0; k < ntt; ++k) {
        int cur = atomicAdd(&cnt->tetWrite, 1);
        long long o = base + (long long)cur * 4;
        for (int c = 0; c < 4; ++c) {
          int e = (int)c_TET[ti][k * 4 + c];
          int id = (e < 4) ? tcomp(t, e) : (N + idxMap[(size_t)idx * 6 + (e - 4)]);
          if (o + c < outCap) ntStore(&out[o + c], (float)remap[id]);
        }
      }
    } else {
      int4 t = innerRow[idx - V];
      int cur = atomicAdd(&cnt->tetWrite, 1);
      long long o = base + (long long)cur * 4;
      if (o + 3 < outCap) {
        ntStore(&out[o + 0], (float)remap[t.x]);
        ntStore(&out[o + 1], (float)remap[t.y]);
        ntStore(&out[o + 2], (float)remap[t.z]);
        ntStore(&out[o + 3], (float)remap[t.w]);
      }
    }
  }
}

// ---------------------------------------------------------------------------
extern "C" void kernel_launch(void* const* d_in, const int* in_sizes, int n_in,
                              void* d_out, int out_size, void* d_ws, size_t ws_size,
                              hipStream_t stream) {
  (void)n_in;
  const float* pos   = (const float*)d_in[0];
  const float* sdf   = (const float*)d_in[1];
  const int*   tet   = (const int*)d_in[2];
  const float* thick = (const float*)d_in[3];
  const int N = in_sizes[0] / 3;
  const int F = in_sizes[2] / 4;

  // ---- workspace partition (256B-aligned) ----
  char* p = (char*)d_ws;
  auto alloc = [&](size_t bytes) -> void* {
    void* r = (void*)p;
    p += (bytes + 255) & ~(size_t)255;
    return r;
  };
  Counters* cnt      = (Counters*)alloc(sizeof(Counters));
  uint32_t* occBits  = (uint32_t*)alloc(((size_t)(N + 31) / 32) * 4);
  int4* validRow     = (int4*)alloc((size_t)F * 16);
  int4* innerRow     = (int4*)alloc((size_t)F * 16);
  int* validTi       = (int*)alloc((size_t)F * 4);
  int* idxMap        = (int*)alloc((size_t)F * 6 * 4);
  int* interpA       = (int*)alloc((size_t)F * 6 * 4);
  int* interpB       = (int*)alloc((size_t)F * 6 * 4);
  size_t flagBytes   = (size_t)N + (size_t)F * 6;
  unsigned char* flags = (unsigned char*)alloc(flagBytes);
  int* remap         = (int*)alloc(flagBytes * 4);
  int* uniqL         = (int*)alloc(flagBytes * 4);
  // hash: largest power of two that fits in the remaining scratch (12B/slot)
  size_t used = (size_t)(p - (char*)d_ws);
  size_t rem  = ws_size > used ? ws_size - used : 0;
  int hbits = 16;
  while (hbits < 25 && ((size_t)12 << (hbits + 1)) <= rem) ++hbits;
  size_t H = (size_t)1 << hbits;
  unsigned hmask = (unsigned)(H - 1);
  unsigned long long* hkey = (unsigned long long*)alloc(H * 8);
  int* hval = (int*)alloc(H * 4);

  // ---- per-call (re)initialization: deterministic across graph replays ----
  hipMemsetAsync(cnt, 0, sizeof(Counters), stream);
  hipMemsetAsync(hkey, 0xFF, H * 8, stream);            // EMPTY_KEY
  hipMemsetAsync(flags, 0, flagBytes, stream);

  const int T = 256;
  const int GS = 2048;                                   // grid-stride kernels
  long long outCap = (long long)out_size;
  float* out = (float*)d_out;
  int clsBlocks = (F + T - 1) / T; if (clsBlocks > 4096) clsBlocks = 4096;

  k_occ<<<(N + T - 1) / T, T, 0, stream>>>(sdf, N, thick, occBits);
  k_classify<<<clsBlocks, T, 0, stream>>>(tet, F, occBits, cnt,
                                          validRow, validTi, innerRow);
  k_edge_insert<<<GS, T, 0, stream>>>(validRow, cnt, hkey, hmask);
  k_edge_assign<<<GS, T, 0, stream>>>(hkey, hval, (unsigned)H, occBits, cnt,
                                      interpA, interpB);
  k_idxmap<<<GS, T, 0, stream>>>(validRow, cnt, hkey, hval, hmask, idxMap);
  k_interp<<<GS, T, 0, stream>>>(pos, sdf, thick, cnt, interpA, interpB, out, outCap);
  k_faces<<<GS, T, 0, stream>>>(cnt, validTi, idxMap, out, outCap);
  k_mark<<<GS, T, 0, stream>>>(cnt, validRow, validTi, innerRow, idxMap, N, flags);
  k_compact<<<GS, T, 0, stream>>>(cnt, flags, remap, uniqL, N);
  k_vtm<<<GS, T, 0, stream>>>(pos, cnt, uniqL, N, out, outCap);
  k_tets<<<GS, T, 0, stream>>>(cnt, validRow, validTi, innerRow, idxMap,
                               remap, N, out, outCap);
}